// BoundaryBCELoss_59021440581903
// MI455X (gfx1250) — compile-verified
//
#include <hip/hip_runtime.h>
#include <math.h>

// Problem constants (from reference): N=64, H=W=384, 5 dilation iterations.
#define IMG_H 384
#define IMG_W 384
#define IMG_N 64
#define ITERS 5

#define TS    64             // output tile (384 = 6*64)
#define RAD   5              // dependence radius of 5 plus-dilations
#define REG   74             // loaded region (TS + 2*RAD)
#define XOFF  3              // left pad: pixel col lx lives at dword XOFF+lx,
                             // so col 1 -> dword 4 (16B aligned for b128)
#define STR   80             // LDS row stride in dwords (mult of 4)
#define CRN   72             // compute region [1,73) in both dims
#define BLK   256
#define NSEG  20             // load segments per row: b32 + 18*b128 + b32
#define NLOADSEG (REG*NSEG)  // 1480
#define NBLK4 (CRN*18)       // 1296 compute blocks of 4 pixels

typedef __attribute__((ext_vector_type(4))) float f4;
typedef __attribute__((ext_vector_type(2))) float v2f;
typedef __attribute__((ext_vector_type(8))) float v8f;

__device__ __forceinline__ unsigned ldsOff(const float* p) {
    // generic pointer to __shared__: low 32 bits = workgroup-relative LDS offset
    return (unsigned)(unsigned long long)p;
}
__device__ __forceinline__ void asyncLoadB32(const float* lds, const float* g) {
    asm volatile("global_load_async_to_lds_b32 %0, %1, off"
                 :: "v"(ldsOff(lds)), "v"(g) : "memory");
}
__device__ __forceinline__ void asyncLoadB128(const float* lds, const float* g) {
    asm volatile("global_load_async_to_lds_b128 %0, %1, off"
                 :: "v"(ldsOff(lds)), "v"(g) : "memory");
}

// -----------------------------------------------------------------------------
// Fused: async global->LDS tile fill (mostly B128), 5 in-LDS dilations per mask
// with 4-pixel register blocking (6 DS ops / 4 px), p=h*o, BCE, deterministic
// block-partial reduction using the V_WMMA_F32_16X16X4_F32 row-sum trick.
// -----------------------------------------------------------------------------
__global__ __launch_bounds__(BLK) void boundary_bce_main(
    const float* __restrict__ hand, const float* __restrict__ obj,
    const float* __restrict__ tgt, float* __restrict__ partial)
{
    __shared__ __align__(16) float bufA[REG * STR];
    __shared__ __align__(16) float bufB[REG * STR];
    __shared__ __align__(16) float hres[TS * TS];
    __shared__ float red[16];

    const int t   = threadIdx.x;
    const int gy0 = blockIdx.y * TS;
    const int gx0 = blockIdx.x * TS;
    const size_t imgBase = (size_t)blockIdx.z * IMG_H * IMG_W;
    const bool edge = (blockIdx.x == 0) | (blockIdx.x == gridDim.x - 1) |
                      (blockIdx.y == 0) | (blockIdx.y == gridDim.y - 1);

    for (int m = 0; m < 2; ++m) {
        const float* src = (m == 0) ? hand : obj;

        // ---- Fill 74x74 region of bufA via async global->LDS ----------------
        for (int i = 0; i < 6; ++i) {
            int s = t + i * BLK;
            if (s < NLOADSEG) {
                int row = s / NSEG;
                int seg = s - row * NSEG;
                int gy  = gy0 - RAD + row;
                bool yin = (unsigned)gy < (unsigned)IMG_H;
                const float* grow = src + imgBase + (size_t)gy * IMG_W;
                float* lrow = bufA + row * STR + XOFF;
                if (seg == 0 || seg == NSEG - 1) {        // cols 0 and 73: b32
                    int lx = (seg == 0) ? 0 : REG - 1;
                    int gx = gx0 - RAD + lx;
                    if (yin & ((unsigned)gx < (unsigned)IMG_W))
                        asyncLoadB32(lrow + lx, grow + gx);
                    else
                        lrow[lx] = 0.0f;
                } else {                                   // cols 1+4k: b128
                    int lx = 1 + 4 * (seg - 1);
                    int gx = gx0 - RAD + lx;               // gx % 4 == 0
                    if (yin && gx >= 0 && gx + 3 < IMG_W) {
                        asyncLoadB128(lrow + lx, grow + gx);
                    } else {
#pragma unroll
                        for (int e = 0; e < 4; ++e) {
                            int gxe = gx + e;
                            if (yin & ((unsigned)gxe < (unsigned)IMG_W))
                                asyncLoadB32(lrow + lx + e, grow + gxe);
                            else
                                lrow[lx + e] = 0.0f;
                        }
                    }
                }
            }
        }
        asm volatile("s_wait_asynccnt 0" ::: "memory");  // per-wave async drain
        __syncthreads();                                 // all waves' data visible

        // ---- 5 dilation iterations, ping-pong, 4-px register blocking -------
        // Fixed compute window rows/cols [1,73): stale-ring garbage marches
        // inward 1/iter from both ends and never reaches final [5,69) window.
        // Out-of-image pixels forced to 0 each iteration (select, NaN-safe).
        for (int it = 0; it < ITERS; ++it) {
            const float* sb = (it & 1) ? bufB : bufA;
            float*       db = (it & 1) ? bufA : bufB;
            for (int i = 0; i < 6; ++i) {
                int idx = t + i * BLK;
                if (idx < NBLK4) {
                    int q   = idx / 18;
                    int ly  = 1 + q;
                    int lxb = 1 + 4 * (idx - q * 18);      // 1,5,...,69
                    int dw  = ly * STR + XOFF + lxb;       // dword, 16B aligned
                    f4 cc = *(const f4*)(sb + dw);
                    f4 up = *(const f4*)(sb + dw - STR);
                    f4 dn = *(const f4*)(sb + dw + STR);
                    float cl = sb[dw - 1];
                    float cr = sb[dw + 4];
                    f4 lf = { cl,   cc.x, cc.y, cc.z };
                    f4 rt = { cc.y, cc.z, cc.w, cr   };
                    f4 sv = cc + up + dn + lf + rt;        // pk_add friendly
                    sv.x = __builtin_amdgcn_fmed3f(sv.x, 0.0f, 1.0f);
                    sv.y = __builtin_amdgcn_fmed3f(sv.y, 0.0f, 1.0f);
                    sv.z = __builtin_amdgcn_fmed3f(sv.z, 0.0f, 1.0f);
                    sv.w = __builtin_amdgcn_fmed3f(sv.w, 0.0f, 1.0f);
                    if (edge) {                            // uniform branch
                        int gy  = gy0 - RAD + ly;
                        int gxb = gx0 - RAD + lxb;
                        bool yin = (unsigned)gy < (unsigned)IMG_H;
                        sv.x = (yin & ((unsigned)(gxb + 0) < (unsigned)IMG_W)) ? sv.x : 0.0f;
                        sv.y = (yin & ((unsigned)(gxb + 1) < (unsigned)IMG_W)) ? sv.y : 0.0f;
                        sv.z = (yin & ((unsigned)(gxb + 2) < (unsigned)IMG_W)) ? sv.z : 0.0f;
                        sv.w = (yin & ((unsigned)(gxb + 3) < (unsigned)IMG_W)) ? sv.w : 0.0f;
                    }
                    *(f4*)(db + dw) = sv;                  // ds_store_b128
                }
            }
            __syncthreads();
        }

        // ITERS=5 is odd -> result in bufB. Stash hand result (float4 copies).
        if (m == 0) {
            for (int i = 0; i < 4; ++i) {
                int f  = t + i * BLK;                      // 0..1023
                int px = f * 4;
                int ly = px >> 6, lx = px & 63;            // lx % 4 == 0
                *(f4*)&hres[px] =
                    *(const f4*)(bufB + (RAD + ly) * STR + XOFF + RAD + lx);
            }
            __syncthreads();   // bufB free before obj pass reuses it
        }
    }

    // ---- p = h*o, BCE terms (log clamped at -100), float4 everywhere --------
    float acc = 0.0f;
    const float* timg = tgt + imgBase;
    for (int i = 0; i < 4; ++i) {              // exact trip count: no divergence
        int f  = t + i * BLK;
        int px = f * 4;
        int ly = px >> 6, lx = px & 63;
        f4 h4 = *(const f4*)&hres[px];
        f4 o4 = *(const f4*)(bufB + (RAD + ly) * STR + XOFF + RAD + lx);
        f4 t4 = *(const f4*)(timg + (size_t)(gy0 + ly) * IMG_W + gx0 + lx);
        f4 p  = h4 * o4;
        acc += t4.x * fmaxf(logf(p.x), -100.0f) + (1.0f - t4.x) * fmaxf(logf(1.0f - p.x), -100.0f);
        acc += t4.y * fmaxf(logf(p.y), -100.0f) + (1.0f - t4.y) * fmaxf(logf(1.0f - p.y), -100.0f);
        acc += t4.z * fmaxf(logf(p.z), -100.0f) + (1.0f - t4.z) * fmaxf(logf(1.0f - p.z), -100.0f);
        acc += t4.w * fmaxf(logf(p.w), -100.0f) + (1.0f - t4.w) * fmaxf(logf(1.0f - p.w), -100.0f);
    }

    // ---- Wave reduction via V_WMMA_F32_16X16X4_F32 --------------------------
    // A(16x4): vgpr0 = per-lane acc (K=0 lanes 0-15, K=2 lanes 16-31), vgpr1=0.
    // B = ones => D[m][n] = acc[m] + acc[m+16]; sum d0..d7 per lane, then
    // lane0 + lane16 hold disjoint half-sums => full 32-lane total.
    // EXEC is all ones here (uniform control flow after barrier).
    v2f a = { acc, 0.0f };
    v2f b = { 1.0f, 1.0f };
    v8f c = { 0.f, 0.f, 0.f, 0.f, 0.f, 0.f, 0.f, 0.f };
    v8f d = __builtin_amdgcn_wmma_f32_16x16x4_f32(false, a, false, b,
                                                  (short)0, c, false, false);
    float s = d[0] + d[1] + d[2] + d[3] + d[4] + d[5] + d[6] + d[7];

    int lane = t & 31, wid = t >> 5;
    if (lane == 0 || lane == 16) red[wid * 2 + (lane >> 4)] = s;
    __syncthreads();
    if (t == 0) {
        float tot = 0.0f;
        for (int i = 0; i < 16; ++i) tot += red[i];   // fixed order: deterministic
        partial[((size_t)blockIdx.z * gridDim.y + blockIdx.y) * gridDim.x
                + blockIdx.x] = tot;
    }
}

// -----------------------------------------------------------------------------
// Deterministic final reduction of block partials -> scalar loss.
// -----------------------------------------------------------------------------
__global__ __launch_bounds__(BLK) void boundary_bce_reduce(
    const float* __restrict__ partial, float* __restrict__ out, int n)
{
    __shared__ float s[BLK];
    float a = 0.0f;
    for (int i = threadIdx.x; i < n; i += BLK) a += partial[i];  // fixed order
    s[threadIdx.x] = a;
    __syncthreads();
    for (int k = BLK / 2; k > 0; k >>= 1) {
        if ((int)threadIdx.x < k) s[threadIdx.x] += s[threadIdx.x + k];
        __syncthreads();
    }
    if (threadIdx.x == 0)
        out[0] = -s[0] * (1.0f / ((float)IMG_N * IMG_H * IMG_W));
}

extern "C" void kernel_launch(void* const* d_in, const int* in_sizes, int n_in,
                              void* d_out, int out_size, void* d_ws, size_t ws_size,
                              hipStream_t stream)
{
    const float* hand = (const float*)d_in[0];
    const float* obj  = (const float*)d_in[1];
    const float* tgt  = (const float*)d_in[2];
    float* partialBuf = (float*)d_ws;          // 6*6*64 = 2304 floats of scratch
    float* out        = (float*)d_out;

    dim3 grid(IMG_W / TS, IMG_H / TS, IMG_N);  // 6 x 6 x 64 = 2304 blocks
    dim3 block(BLK);
    boundary_bce_main<<<grid, block, 0, stream>>>(hand, obj, tgt, partialBuf);
    boundary_bce_reduce<<<1, BLK, 0, stream>>>(partialBuf, out,
                                               (IMG_W / TS) * (IMG_H / TS) * IMG_N);
}